// GATLayer_53274774340076
// MI455X (gfx1250) — compile-verified
//
#include <hip/hip_runtime.h>
#include <hip/hip_bf16.h>

#define NIN   128          // input features (= K of GEMM)
#define NOUT  128          // HEADS*HID
#define HEADS 8
#define HID   16
#define NEG_SLOPE 0.2f

typedef __attribute__((ext_vector_type(2))) float v2f;
typedef __attribute__((ext_vector_type(8))) float v8f;

// ---------------------------------------------------------------------------
// Kernel 0: zero a float buffer (harness poisons d_out / d_ws with 0xAA)
// ---------------------------------------------------------------------------
__global__ void gat_zero(float* __restrict__ p, int n) {
    int i = blockIdx.x * blockDim.x + threadIdx.x;
    if (i < n) p[i] = 0.0f;
}

// ---------------------------------------------------------------------------
// Kernel 1: h = x @ W^T via V_WMMA_F32_16X16X4_F32, fused attention logits.
//   block = 256 threads = 8 waves; block handles 16 rows of x.
//   wave w computes the 16x16 tile for head w (cols 16w..16w+15).
//   A fragment (16x4 f32): lane m_lo holds x[row0+m_lo, k+2*quad+{0,1}]
//   B fragment (4x16 f32): lane m_lo holds W[col0+m_lo, k+2*quad+{0,1}]
//   D tile: lane(m_lo,quad) vgpr v = h[row0+v+8*quad, col0+m_lo]
// ---------------------------------------------------------------------------
__global__ __launch_bounds__(256) void gat_gemm_wmma(
    const float* __restrict__ x, const float* __restrict__ W,
    const float* __restrict__ att,
    float* __restrict__ h, float* __restrict__ al, float* __restrict__ ar)
{
    const int lane = threadIdx.x & 31;
    const int wv   = threadIdx.x >> 5;     // wave id == head id
    const int row0 = blockIdx.x * 16;
    const int m_lo = lane & 15;
    const int quad = lane >> 4;            // 0 or 1

    const float* ax = x + (size_t)(row0 + m_lo) * NIN + 2 * quad;
    const float* bw = W + (size_t)(wv * 16 + m_lo) * NIN + 2 * quad;

    v8f c = {0.f, 0.f, 0.f, 0.f, 0.f, 0.f, 0.f, 0.f};
#pragma unroll
    for (int kk = 0; kk < NIN / 4; ++kk) {
        v2f a = *(const v2f*)(ax + kk * 4);
        v2f b = *(const v2f*)(bw + kk * 4);
        c = __builtin_amdgcn_wmma_f32_16x16x4_f32(
                /*neg_a=*/false, a, /*neg_b=*/false, b,
                /*c_mod=*/(short)0, c, /*reuse_a=*/false, /*reuse_b=*/false);
    }

    // store h tile
#pragma unroll
    for (int v = 0; v < 8; ++v) {
        const int node = row0 + v + 8 * quad;
        h[(size_t)node * NOUT + wv * HID + m_lo] = c[v];
    }

    // fused attention logits: al[node,w] = sum_d h[node,w,d]*att_l[w,d]
    // d == m_lo within the wave's tile -> reduce across 16-lane N-groups.
    const float attl = att[wv * (2 * HID) + m_lo];
    const float attr = att[wv * (2 * HID) + HID + m_lo];
#pragma unroll
    for (int v = 0; v < 8; ++v) {
        float sl = c[v] * attl;
        float sr = c[v] * attr;
#pragma unroll
        for (int m = 8; m >= 1; m >>= 1) {
            sl += __shfl_xor(sl, m, 16);
            sr += __shfl_xor(sr, m, 16);
        }
        if (m_lo == 0) {
            const int node = row0 + v + 8 * quad;
            al[node * HEADS + wv] = sl;
            ar[node * HEADS + wv] = sr;
        }
    }
}

// ---------------------------------------------------------------------------
// Kernel 2: per-edge alpha = exp(leaky(al[row]+ar[col])) * pos[col];
//           rowsum[row,h] += alpha  (hardware f32 atomics)
// ---------------------------------------------------------------------------
__global__ __launch_bounds__(256) void gat_edge_alpha(
    const int* __restrict__ row, const int* __restrict__ col,
    const float* __restrict__ al, const float* __restrict__ ar,
    const float* __restrict__ pos,
    float* __restrict__ alpha, float* __restrict__ rsum, int E_)
{
    const int e = blockIdx.x * blockDim.x + threadIdx.x;
    if (e >= E_) return;
    const int r = row[e];
    const int c = col[e];
    const float p = pos[c];

    const float4 l0 = *(const float4*)(al + (size_t)r * HEADS);
    const float4 l1 = *(const float4*)(al + (size_t)r * HEADS + 4);
    const float4 r0 = *(const float4*)(ar + (size_t)c * HEADS);
    const float4 r1 = *(const float4*)(ar + (size_t)c * HEADS + 4);

    float av[HEADS] = {l0.x + r0.x, l0.y + r0.y, l0.z + r0.z, l0.w + r0.w,
                       l1.x + r1.x, l1.y + r1.y, l1.z + r1.z, l1.w + r1.w};
    float o[HEADS];
#pragma unroll
    for (int hh = 0; hh < HEADS; ++hh) {
        float a = av[hh];
        a = (a >= 0.0f) ? a : NEG_SLOPE * a;
        a = __expf(a) * p;
        o[hh] = a;
        unsafeAtomicAdd(&rsum[(size_t)r * HEADS + hh], a);
    }
    *(float4*)(alpha + (size_t)e * HEADS)     = make_float4(o[0], o[1], o[2], o[3]);
    *(float4*)(alpha + (size_t)e * HEADS + 4) = make_float4(o[4], o[5], o[6], o[7]);
}

// ---------------------------------------------------------------------------
// Kernel 3: rsum -> inv = 1/rsum + 1e-16 (in place)
// ---------------------------------------------------------------------------
__global__ void gat_inv(float* __restrict__ rsum, int n) {
    int i = blockIdx.x * blockDim.x + threadIdx.x;
    if (i < n) rsum[i] = 1.0f / rsum[i] + 1e-16f;
}

// ---------------------------------------------------------------------------
// Kernel 4: out[row] += (alpha[e]*inv[row]) * h[col]
//   one wave per edge; lane l owns 4 contiguous floats (head = l/4).
// ---------------------------------------------------------------------------
__global__ __launch_bounds__(256) void gat_aggregate(
    const int* __restrict__ row, const int* __restrict__ col,
    const float* __restrict__ alpha, const float* __restrict__ inv,
    const float* __restrict__ h, float* __restrict__ out, int E_)
{
    const int lane = threadIdx.x & 31;
    const int e = (int)((blockIdx.x * (size_t)blockDim.x + threadIdx.x) >> 5);
    if (e >= E_) return;
    const int r = row[e];
    const int c = col[e];
    const int head = lane >> 2;                         // 4 floats per lane, 16 per head

    const float scale = alpha[(size_t)e * HEADS + head] * inv[(size_t)r * HEADS + head];
    const float4 hv = *(const float4*)(h + (size_t)c * NOUT + lane * 4);
    float* dst = out + (size_t)r * NOUT + lane * 4;
    unsafeAtomicAdd(dst + 0, scale * hv.x);
    unsafeAtomicAdd(dst + 1, scale * hv.y);
    unsafeAtomicAdd(dst + 2, scale * hv.z);
    unsafeAtomicAdd(dst + 3, scale * hv.w);
}

// ---------------------------------------------------------------------------
extern "C" void kernel_launch(void* const* d_in, const int* in_sizes, int n_in,
                              void* d_out, int out_size, void* d_ws, size_t ws_size,
                              hipStream_t stream) {
    const float* x   = (const float*)d_in[0];   // N x 128
    const int*   ei  = (const int*)  d_in[1];   // 2 x E
    const float* pos = (const float*)d_in[2];   // N
    const float* W   = (const float*)d_in[3];   // 128 x 128
    const float* att = (const float*)d_in[4];   // 1 x 8 x 32

    float* out = (float*)d_out;                 // N x 128

    const int N_ = in_sizes[0] / NIN;           // 100000
    const int E_ = in_sizes[1] / 2;             // 1600000

    const int* row = ei;
    const int* col = ei + E_;

    // workspace layout (floats): h[N*128] | al[N*8] | ar[N*8] | rsum[N*8] | alpha[E*8]
    float* ws    = (float*)d_ws;
    float* h     = ws;
    float* al    = h    + (size_t)N_ * NOUT;
    float* ar    = al   + (size_t)N_ * HEADS;
    float* rsum  = ar   + (size_t)N_ * HEADS;
    float* alpha = rsum + (size_t)N_ * HEADS;

    // zero accumulation buffers
    {
        int n = N_ * NOUT;
        gat_zero<<<(n + 255) / 256, 256, 0, stream>>>(out, n);
        n = N_ * HEADS;
        gat_zero<<<(n + 255) / 256, 256, 0, stream>>>(rsum, n);
    }

    // 1: WMMA GEMM + fused logits  (N divisible by 16: 100000/16 = 6250)
    gat_gemm_wmma<<<N_ / 16, 256, 0, stream>>>(x, W, att, h, al, ar);

    // 2: edge softmax numerators + row sums
    gat_edge_alpha<<<(E_ + 255) / 256, 256, 0, stream>>>(row, col, al, ar, pos,
                                                         alpha, rsum, E_);

    // 3: reciprocal row sums
    gat_inv<<<(N_ * HEADS + 255) / 256, 256, 0, stream>>>(rsum, N_ * HEADS);

    // 4: weighted aggregation, one wave per edge
    gat_aggregate<<<(E_ + 7) / 8, 256, 0, stream>>>(row, col, alpha, rsum, h,
                                                    out, E_);
}